// GraphAttentionLayer_61435212202106
// MI455X (gfx1250) — compile-verified
//
#include <hip/hip_runtime.h>
#include <cstddef>
#include <cstdint>

// Problem constants (from reference setup_inputs)
static constexpr int kN = 8;     // batch
static constexpr int kV = 2048;  // nodes
static constexpr int kC = 128;   // in features
static constexpr int kE = 64;    // out features
#define LRELU_ALPHA 0.2f

typedef __attribute__((ext_vector_type(2))) float v2f;
typedef __attribute__((ext_vector_type(8))) float v8f;

__device__ __forceinline__ v8f wmma_f32_16x16x4(v2f a, v2f b, v8f c) {
    // D = A(16x4 f32) * B(4x16 f32) + C(16x16 f32)
    return __builtin_amdgcn_wmma_f32_16x16x4_f32(
        /*neg_a=*/false, a, /*neg_b=*/false, b,
        /*c_mod=*/(short)0, c, /*reuse_a=*/false, /*reuse_b=*/false);
}

__device__ __forceinline__ float lrelu(float x) {
    return x > 0.0f ? x : LRELU_ALPHA * x;
}

// Order-preserving uint key for float min/max via integer atomics
__device__ __forceinline__ unsigned fkey(float x) {
    unsigned u = __float_as_uint(x);
    return (u & 0x80000000u) ? ~u : (u | 0x80000000u);
}
__device__ __forceinline__ float funkey(unsigned k) {
    return (k & 0x80000000u) ? __uint_as_float(k & 0x7FFFFFFFu)
                             : __uint_as_float(~k);
}

// Async copy of a contiguous 16KB tile (64x64 f32) global -> LDS.
// gfx1250 GLOBAL_LOAD_ASYNC_TO_LDS_B128 (ASYNCcnt): INST_OFFSET is added to
// BOTH the LDS and global addresses, so a single per-lane (lds, goff) pair
// plus 8 issues covers the tile. 128 threads x 8 x 16B = 16384B.
__device__ __forceinline__ void async_tile_16k(unsigned lds_base,
                                               const float* __restrict__ gsrc,
                                               int tid) {
    unsigned lds  = lds_base + (unsigned)tid * 16u;
    unsigned goff = (unsigned)tid * 16u;
    #pragma unroll
    for (int q = 0; q < 8; ++q) {
        asm volatile("global_load_async_to_lds_b128 %0, %1, %2"
                     :: "v"(lds + (unsigned)(q * 2048)),
                        "v"(goff + (unsigned)(q * 2048)),
                        "s"(gsrc)
                     : "memory");
    }
}
__device__ __forceinline__ void wait_async0() {
    asm volatile("s_wait_asynccnt 0" ::: "memory");
}

// ---------------- K0: init stats ----------------
__global__ void k_init(unsigned* stats) {
    if (threadIdx.x == 0) { stats[0] = 0xFFFFFFFFu; stats[1] = 0u; }
}

// ---------------- K1: rowsum(B) + min/max(B + I) ----------------
__global__ __launch_bounds__(256) void k_adj_stats(const float* __restrict__ Bm,
                                                   unsigned* __restrict__ stats,
                                                   float* __restrict__ rowsum) {
    const int u = blockIdx.x;
    const int t = threadIdx.x;
    float s = 0.0f, mn = 3.402823466e+38f, mx = -3.402823466e+38f;
    for (int v = t; v < kV; v += 256) {
        float b = Bm[(size_t)u * kV + v];
        s += b;
        float adj = b + (v == u ? 1.0f : 0.0f);
        mn = fminf(mn, adj);
        mx = fmaxf(mx, adj);
    }
    __shared__ float ss[256], smn[256], smx[256];
    ss[t] = s; smn[t] = mn; smx[t] = mx;
    __syncthreads();
    for (int w = 128; w >= 1; w >>= 1) {
        if (t < w) {
            ss[t] += ss[t + w];
            smn[t] = fminf(smn[t], smn[t + w]);
            smx[t] = fmaxf(smx[t], smx[t + w]);
        }
        __syncthreads();
    }
    if (t == 0) {
        rowsum[u] = ss[0];
        atomicMin(&stats[0], fkey(smn[0]));
        atomicMax(&stats[1], fkey(smx[0]));
    }
}

// ---------------- K2: d12[u] = rsqrt(deg), stash mn/invrange ----------------
__global__ void k_d12(const unsigned* __restrict__ stats,
                      const float* __restrict__ rowsum,
                      float* __restrict__ d12, float* __restrict__ scal) {
    const float mn = funkey(stats[0]);
    const float mx = funkey(stats[1]);
    const float invr = 1.0f / (mx - mn);
    const int u = blockIdx.x * blockDim.x + threadIdx.x;
    if (u < kV) {
        float deg = invr * (rowsum[u] + 1.0f - (float)kV * mn);
        d12[u] = rsqrtf(deg);
    }
    if (u == 0) { scal[0] = mn; scal[1] = invr; }
}

// ---------------- K3: Wh = h@W  (fused f1 = Wh@a1, f2 = Wh@a2) ----------------
__global__ __launch_bounds__(64) void k_wh(const float* __restrict__ h,
                                           const float* __restrict__ W,
                                           const float* __restrict__ a,
                                           float* __restrict__ Wh,
                                           float* __restrict__ f1,
                                           float* __restrict__ f2) {
    const int nv = blockIdx.x;   // 0 .. N*V-1
    const int e  = threadIdx.x;  // 0 .. 63
    __shared__ float hs[kC];
    __shared__ float red[64];
    hs[e]      = h[(size_t)nv * kC + e];
    hs[e + 64] = h[(size_t)nv * kC + 64 + e];
    __syncthreads();
    float acc = 0.0f;
    #pragma unroll 8
    for (int c = 0; c < kC; ++c)
        acc = fmaf(hs[c], W[c * kE + e], acc);
    Wh[(size_t)nv * kE + e] = acc;
    red[e] = acc * a[e];
    __syncthreads();
    for (int w = 32; w >= 1; w >>= 1) {
        if (e < w) red[e] += red[e + w];
        __syncthreads();
    }
    if (e == 0) f1[nv] = red[0];
    __syncthreads();
    red[e] = acc * a[kE + e];
    __syncthreads();
    for (int w = 32; w >= 1; w >>= 1) {
        if (e < w) red[e] += red[e + w];
        __syncthreads();
    }
    if (e == 0) f2[nv] = red[0];
}

// ---------------- K3b: m2[n] = max_v f2[n,v] ----------------
__global__ __launch_bounds__(256) void k_m2(const float* __restrict__ f2,
                                            float* __restrict__ m2) {
    const int n = blockIdx.x, t = threadIdx.x;
    float m = -3.402823466e+38f;
    for (int v = t; v < kV; v += 256) m = fmaxf(m, f2[(size_t)n * kV + v]);
    __shared__ float sm[256];
    sm[t] = m;
    __syncthreads();
    for (int w = 128; w >= 1; w >>= 1) {
        if (t < w) sm[t] = fmaxf(sm[t], sm[t + w]);
        __syncthreads();
    }
    if (t == 0) m2[n] = sm[0];
}

// ---------------- K4: Y[n] = softmax(lrelu(f1+f2)) @ Wh[n] ----------------
// Flash-style: fixed row-max = lrelu(f1[u]+max(f2)), P tiles never hit global.
// Double-buffered: async-to-LDS prefetch of Wh chunk i+1 overlaps WMMA on i.
__global__ __launch_bounds__(128) void k_attn(const float* __restrict__ f1,
                                              const float* __restrict__ f2,
                                              const float* __restrict__ m2,
                                              const float* __restrict__ Wh,
                                              float* __restrict__ Y) {
    const int n  = blockIdx.y;
    const int u0 = blockIdx.x * 16;
    const int tid  = threadIdx.x;
    const int wave = tid >> 5;       // 4 waves: e-columns wave*16 .. +15
    const int lane = tid & 31;
    __shared__ float P[2][16][65];   // prob tile (padded), double buffered
    __shared__ float Whs[2][64][64]; // Wh chunk: v x e, double buffered
    __shared__ float Zred[16][8];
    __shared__ float Zsum[16];
    const unsigned whs_base = (unsigned)(uintptr_t)(&Whs[0][0][0]);

    const int pi  = tid >> 3;        // row this thread fills (0..15)
    const int pj0 = (tid & 7) * 8;   // 8 columns within chunk
    const float f1u  = f1[(size_t)n * kV + u0 + pi];
    const float Mrow = lrelu(f1u + m2[n]);
    float zpart = 0.0f;

    v8f acc = {0.f, 0.f, 0.f, 0.f, 0.f, 0.f, 0.f, 0.f};
    const int e0    = wave * 16;
    const int mrow  = lane & 15;
    const int khalf = (lane >> 4) * 2;  // K split per ISA f32 A/B layout
    const float* whn = Wh + (size_t)n * kV * kE;
    const float* f2n = f2 + (size_t)n * kV;

    // Prologue: stage chunk 0 into buffer 0
    async_tile_16k(whs_base, whn, tid);
    #pragma unroll
    for (int jj = 0; jj < 8; ++jj) {
        float s = lrelu(f1u + f2n[pj0 + jj]);
        float p = __expf(s - Mrow);
        P[0][pi][pj0 + jj] = p;
        zpart += p;
    }
    wait_async0();
    __syncthreads();

    for (int i = 0; i < kV / 64; ++i) {
        const int cur = i & 1, nxt = cur ^ 1;
        if (i + 1 < kV / 64) {
            const int v1 = (i + 1) * 64;
            // Prefetch next Wh chunk straight into LDS (ASYNCcnt)
            async_tile_16k(whs_base + (unsigned)nxt * 16384u, whn + (size_t)v1 * kE, tid);
            // Build next P tile (VALU/TRANS co-executes with WMMA below)
            #pragma unroll
            for (int jj = 0; jj < 8; ++jj) {
                const int v = v1 + pj0 + jj;
                float s = lrelu(f1u + f2n[v]);
                float p = __expf(s - Mrow);
                P[nxt][pi][pj0 + jj] = p;
                zpart += p;
            }
        }
        // 16 WMMAs over K=64 on the current buffers
        #pragma unroll
        for (int k = 0; k < 64; k += 4) {
            v2f A;  A.x  = P[cur][mrow][k + khalf];
                    A.y  = P[cur][mrow][k + khalf + 1];
            v2f Bv; Bv.x = Whs[cur][k + khalf][e0 + mrow];
                    Bv.y = Whs[cur][k + khalf + 1][e0 + mrow];
            acc = wmma_f32_16x16x4(A, Bv, acc);
        }
        wait_async0();
        __syncthreads();
    }
    // Row-wise softmax denominators
    Zred[pi][tid & 7] = zpart;
    __syncthreads();
    if (tid < 16) {
        float z = 0.0f;
        #pragma unroll
        for (int q = 0; q < 8; ++q) z += Zred[tid][q];
        Zsum[tid] = z;
    }
    __syncthreads();
    // Scale + store: vgpr r holds row m = r + 8*(lane>=16), col = lane%16
    const int ncol = lane & 15, half = lane >> 4;
    #pragma unroll
    for (int r = 0; r < 8; ++r) {
        const int m = r + half * 8;
        Y[((size_t)n * kV + u0 + m) * kE + e0 + ncol] = acc[r] / Zsum[m];
    }
}

// ---------------- K5: out = elu( adj_norm @ Y ) ----------------
__global__ __launch_bounds__(128) void k_final(const float* __restrict__ Bm,
                                               const float* __restrict__ d12,
                                               const float* __restrict__ scal,
                                               const float* __restrict__ Y,
                                               float* __restrict__ out) {
    const int n  = blockIdx.y;
    const int u0 = blockIdx.x * 16;
    const int tid  = threadIdx.x;
    const int wave = tid >> 5;
    const int lane = tid & 31;
    const float mn = scal[0], invr = scal[1];
    __shared__ float G[2][16][65];   // (B[u,v]+delta-mn)*d12[v], double buffered
    __shared__ float Ys[2][64][64];  // Y chunk: v x e, double buffered
    const unsigned ys_base = (unsigned)(uintptr_t)(&Ys[0][0][0]);

    const int pi  = tid >> 3;
    const int pj0 = (tid & 7) * 8;
    v8f acc = {0.f, 0.f, 0.f, 0.f, 0.f, 0.f, 0.f, 0.f};
    const int e0    = wave * 16;
    const int mrow  = lane & 15;
    const int khalf = (lane >> 4) * 2;
    const float* yn   = Y + (size_t)n * kV * kE;
    const float* brow = Bm + (size_t)(u0 + pi) * kV;

    // Prologue: chunk 0 into buffer 0
    async_tile_16k(ys_base, yn, tid);
    #pragma unroll
    for (int jj = 0; jj < 8; ++jj) {
        const int v = pj0 + jj;
        float g = brow[v] + ((u0 + pi) == v ? 1.0f : 0.0f) - mn;
        G[0][pi][pj0 + jj] = g * d12[v];
    }
    wait_async0();
    __syncthreads();

    for (int i = 0; i < kV / 64; ++i) {
        const int cur = i & 1, nxt = cur ^ 1;
        if (i + 1 < kV / 64) {
            const int v1 = (i + 1) * 64;
            async_tile_16k(ys_base + (unsigned)nxt * 16384u, yn + (size_t)v1 * kE, tid);
            #pragma unroll
            for (int jj = 0; jj < 8; ++jj) {
                const int v = v1 + pj0 + jj;
                float g = brow[v] + ((u0 + pi) == v ? 1.0f : 0.0f) - mn;
                G[nxt][pi][pj0 + jj] = g * d12[v];
            }
        }
        #pragma unroll
        for (int k = 0; k < 64; k += 4) {
            v2f A;  A.x  = G[cur][mrow][k + khalf];
                    A.y  = G[cur][mrow][k + khalf + 1];
            v2f Bv; Bv.x = Ys[cur][k + khalf][e0 + mrow];
                    Bv.y = Ys[cur][k + khalf + 1][e0 + mrow];
            acc = wmma_f32_16x16x4(A, Bv, acc);
        }
        wait_async0();
        __syncthreads();
    }
    const int ncol = lane & 15, half = lane >> 4;
    #pragma unroll
    for (int r = 0; r < 8; ++r) {
        const int m = r + half * 8;
        const int u = u0 + m;
        float x = acc[r] * d12[u] * invr;             // left scale d12[u]*invr
        float o = x > 0.0f ? x : (__expf(x) - 1.0f);  // ELU (alpha=1)
        out[((size_t)n * kV + u) * kE + e0 + ncol] = o;
    }
}

extern "C" void kernel_launch(void* const* d_in, const int* in_sizes, int n_in,
                              void* d_out, int out_size, void* d_ws, size_t ws_size,
                              hipStream_t stream) {
    const float* h  = (const float*)d_in[0];  // (N,V,C)
    const float* W  = (const float*)d_in[1];  // (C,E)
    const float* a  = (const float*)d_in[2];  // (2E,1)
    const float* Bm = (const float*)d_in[3];  // (V,V)
    float* out = (float*)d_out;               // (N,V,E)

    // Workspace partition (floats); all float4-aligned
    float* wsf      = (float*)d_ws;
    unsigned* stats = (unsigned*)wsf;              // [0]=min key, [1]=max key
    float* scal     = wsf + 2;                     // mn, invrange
    float* rowsum   = wsf + 16;                    // V
    float* d12      = rowsum + kV;                 // V
    float* f1       = d12 + kV;                    // N*V
    float* f2       = f1 + (size_t)kN * kV;        // N*V
    float* m2       = f2 + (size_t)kN * kV;        // N (pad 16)
    float* Wh       = m2 + 16;                     // N*V*E
    float* Y        = Wh + (size_t)kN * kV * kE;   // N*V*E

    k_init<<<1, 1, 0, stream>>>(stats);
    k_adj_stats<<<kV, 256, 0, stream>>>(Bm, stats, rowsum);
    k_d12<<<kV / 256, 256, 0, stream>>>(stats, rowsum, d12, scal);
    k_wh<<<kN * kV, 64, 0, stream>>>(h, W, a, Wh, f1, f2);
    k_m2<<<kN, 256, 0, stream>>>(f2, m2);
    k_attn<<<dim3(kV / 16, kN), 128, 0, stream>>>(f1, f2, m2, Wh, Y);
    k_final<<<dim3(kV / 16, kN), 128, 0, stream>>>(Bm, d12, scal, Y, out);
}